// CRF_9912784519260
// MI455X (gfx1250) — compile-verified
//
#include <hip/hip_runtime.h>

typedef __attribute__((ext_vector_type(2))) float v2f;
typedef __attribute__((ext_vector_type(8))) float v8f;

#define B_  128
#define T_  256
#define H_  768
#define L_  9
#define LP  16   // padded logits row stride (floats)

// ------------------------------------------------------------------
// Kernel 1: logits = x @ W + b   via V_WMMA_F32_16X16X4_F32 (exact fp32)
// One wave per 16-row output tile; N padded 9 -> 16; K=768 in steps of 4,
// two accumulators interleaved to break the WMMA D->C dependency chain.
// Bandwidth-bound: 100.7 MB of x is the cost; W staged in LDS (48 KB).
// ------------------------------------------------------------------
__global__ __launch_bounds__(256) void gemm_logits_kernel(
    const float* __restrict__ x, const float* __restrict__ W,
    const float* __restrict__ bias, float* __restrict__ logits)
{
  __shared__ float s_w[H_ * LP];   // 48 KB, cols 9..15 zero-padded
  __shared__ float s_b[LP];

  for (int i = threadIdx.x; i < H_ * LP; i += 256) {
    int k = i >> 4, n = i & 15;
    s_w[i] = (n < L_) ? W[k * L_ + n] : 0.0f;
  }
  if (threadIdx.x < LP)
    s_b[threadIdx.x] = (threadIdx.x < L_) ? bias[threadIdx.x] : 0.0f;
  __syncthreads();

  const int lane = threadIdx.x & 31;
  const int wv   = threadIdx.x >> 5;
  const int tile = blockIdx.x * 8 + wv;     // 2048 tiles of 16 rows
  const int row_base = tile * 16;
  const int hi = lane >> 4;                 // lane half selects K+0/1 vs K+2/3
  const int n  = lane & 15;                 // output column / A row

  // A 16x4 f32 layout: lanes 0-15 hold row M=lane, K={k,k+1};
  //                    lanes 16-31 hold row M=lane-16, K={k+2,k+3}.
  const float* xr = x + (size_t)(row_base + n) * H_ + 2 * hi;

  v8f acc0 = {}; v8f acc1 = {};
  for (int k = 0; k < H_; k += 8) {
    v2f a0 = *(const v2f*)(xr + k);
    v2f b0;
    b0.x = s_w[(k     + 2 * hi) * LP + n];  // B 4x16: VGPR0 = rows K=k | K=k+2
    b0.y = s_w[(k + 1 + 2 * hi) * LP + n];  //         VGPR1 = rows K=k+1 | K=k+3
    acc0 = __builtin_amdgcn_wmma_f32_16x16x4_f32(
        false, a0, false, b0, (short)0, acc0, false, false);

    v2f a1 = *(const v2f*)(xr + k + 4);
    v2f b1;
    b1.x = s_w[(k + 4 + 2 * hi) * LP + n];
    b1.y = s_w[(k + 5 + 2 * hi) * LP + n];
    acc1 = __builtin_amdgcn_wmma_f32_16x16x4_f32(
        false, a1, false, b1, (short)0, acc1, false, false);
  }

  // D layout: VGPR v, lanes 0-15 -> M=v, lanes 16-31 -> M=v+8; N = lane&15.
  float bb = s_b[n];
#pragma unroll
  for (int v = 0; v < 8; ++v) {
    int row = row_base + v + 8 * hi;
    logits[(size_t)row * LP + n] = acc0[v] + acc1[v] + bb;
  }
}

// ------------------------------------------------------------------
// Kernel 2: CRF forward (logZ), gold-sequence score, Viterbi + backtrack,
// and loss/accuracy partial sums. One wave per batch row (latency-bound
// sequential scan). Backpointers kept in LDS so the serial backtrack
// chain hits LDS latency, not HBM latency.
// ------------------------------------------------------------------
__global__ __launch_bounds__(128) void crf_kernel(
    const float* __restrict__ logits, const int* __restrict__ label,
    const int* __restrict__ seqlen, const float* __restrict__ trans,
    float* __restrict__ out, float* __restrict__ accum)
{
  __shared__ float s_trans[L_ * L_];
  __shared__ unsigned char s_bp[4][T_][12];   // per-wave backpointers

  if (threadIdx.x < L_ * L_) s_trans[threadIdx.x] = trans[threadIdx.x];
  __syncthreads();

  const int lane = threadIdx.x & 31;
  const int w    = threadIdx.x >> 5;
  const int b    = blockIdx.x * 4 + w;
  const int slen = seqlen[b];
  const int j    = (lane < L_) ? lane : 0;    // this lane's label column

  const float* lg = logits + (size_t)b * T_ * LP;
  const int*   lb = label + b * T_;

  float tc[L_];                                // trans[:, j] in registers
#pragma unroll
  for (int i = 0; i < L_; ++i) tc[i] = s_trans[i * L_ + j];

  float a[L_], va[L_];                         // alpha replicated per lane
#pragma unroll
  for (int i = 0; i < L_; ++i) { a[i] = lg[i]; va[i] = a[i]; }

  int lab_prev = lb[0];
  float score = lg[lab_prev];                  // unary t=0 (always in-mask)

  for (int t = 1; t < T_; ++t) {
    float logit_j = lg[t * LP + j];
    bool  m = (t < slen);
    int   lab = lb[t];

    // log-norm step: new_j = logsumexp_i(a_i + trans[i][j]) + logit_j
    float mx = a[0] + tc[0];
#pragma unroll
    for (int i = 1; i < L_; ++i) mx = fmaxf(mx, a[i] + tc[i]);
    float se = 0.f;
#pragma unroll
    for (int i = 0; i < L_; ++i) se += __expf(a[i] + tc[i] - mx);
    float newv = mx + __logf(se) + logit_j;

    // viterbi step: max/argmax (first-max ties like jnp.argmax)
    float bv = va[0] + tc[0]; int bi = 0;
#pragma unroll
    for (int i = 1; i < L_; ++i) {
      float v2 = va[i] + tc[i];
      if (v2 > bv) { bv = v2; bi = i; }
    }
    float newva = bv + logit_j;

    // gold-path score (lane 0)
    if (lane == 0 && m) score += lg[t * LP + lab] + s_trans[lab_prev * L_ + lab];
    lab_prev = lab;

    // broadcast new alphas across the wave, apply mask (carry if !m)
#pragma unroll
    for (int i = 0; i < L_; ++i) {
      float na  = __shfl(newv,  i, 32);
      float nva = __shfl(newva, i, 32);
      if (m) { a[i] = na; va[i] = nva; }
    }
    if (lane < L_) s_bp[w][t][lane] = (unsigned char)(m ? bi : lane);
  }

  // logZ = logsumexp(alpha_T); last = argmax(valpha_T)  (replicated values)
  float mx = a[0];
#pragma unroll
  for (int i = 1; i < L_; ++i) mx = fmaxf(mx, a[i]);
  float se = 0.f;
#pragma unroll
  for (int i = 0; i < L_; ++i) se += __expf(a[i] - mx);
  float logZ = mx + __logf(se);
  float bvv = va[0]; int last = 0;
#pragma unroll
  for (int i = 1; i < L_; ++i) if (va[i] > bvv) { bvv = va[i]; last = i; }

  if (lane == 0) {
    atomicAdd(&accum[0], logZ - score);        // sum of -log_likelihood
    atomicAdd(&accum[2], (float)slen);         // sum of seqlen
    int tag = last;
    int correct = ((T_ - 1) < slen && tag == lb[T_ - 1]) ? 1 : 0;
    out[b * T_ + (T_ - 1)] = (float)tag;
    for (int t = T_ - 2; t >= 0; --t) {        // backtrack through LDS bps
      tag = s_bp[w][t + 1][tag];
      out[b * T_ + t] = (float)tag;
      if (t < slen && tag == lb[t]) correct++;
    }
    atomicAdd(&accum[1], (float)correct);
  }
}

// ------------------------------------------------------------------
__global__ void init_kernel(float* accum) {
  if (threadIdx.x < 4) accum[threadIdx.x] = 0.f;
}

__global__ void finalize_kernel(const float* __restrict__ accum,
                                float* __restrict__ out) {
  if (threadIdx.x == 0) {
    out[B_ * T_]     = accum[0] / (float)B_;   // loss = mean(-ll)
    out[B_ * T_ + 1] = accum[1] / accum[2];    // accuracy
  }
}

// ------------------------------------------------------------------
extern "C" void kernel_launch(void* const* d_in, const int* in_sizes, int n_in,
                              void* d_out, int out_size, void* d_ws, size_t ws_size,
                              hipStream_t stream) {
  const float* x      = (const float*)d_in[0];
  const int*   label  = (const int*)  d_in[1];
  const int*   seqlen = (const int*)  d_in[2];
  const float* W      = (const float*)d_in[3];
  const float* bias   = (const float*)d_in[4];
  const float* trans  = (const float*)d_in[5];
  float* out = (float*)d_out;

  float* accum  = (float*)d_ws;                         // 4 floats
  float* logits = (float*)((char*)d_ws + 256);          // 32768 x 16 f32 (2 MB)

  hipLaunchKernelGGL(init_kernel, dim3(1), dim3(32), 0, stream, accum);
  hipLaunchKernelGGL(gemm_logits_kernel, dim3(256), dim3(256), 0, stream,
                     x, W, bias, logits);
  hipLaunchKernelGGL(crf_kernel, dim3(32), dim3(128), 0, stream,
                     logits, label, seqlen, trans, out, accum);
  hipLaunchKernelGGL(finalize_kernel, dim3(1), dim3(32), 0, stream, accum, out);
}